// GraphAttention_81965155877400
// MI455X (gfx1250) — compile-verified
//
#include <hip/hip_runtime.h>

typedef float v2f __attribute__((ext_vector_type(2)));
typedef float v8f __attribute__((ext_vector_type(8)));

#define F_DIM 256
#define U_DIM 128
#define DEG   16
#define KC    64      // K-chunk staged in LDS
#define WAVES 8       // waves per workgroup in GEMM

// -------------------------------------------------------------------------
// Kernel 1: h[N,128] = ns[N,256] @ kern[256,128], fp32 WMMA 16x16x4.
// Each wave computes a 16-row x 128-col strip (8 C tiles, 64 accum VGPRs).
// B is staged into LDS interleaved by K-pairs so every B fragment is one
// aligned ds_load_b64 straight into an even VGPR pair (no repack movs).
// -------------------------------------------------------------------------
__global__ __launch_bounds__(256) void gat_gemm_h(
    const float* __restrict__ ns, const float* __restrict__ kern,
    float* __restrict__ h, int N)
{
  // ldsB layout: pair p=(k/2), entry (p*128+col) is float2 {B[k][col],B[k+1][col]}
  __shared__ float ldsB[KC * U_DIM];        // 32 KB
  __shared__ float ldsA[WAVES][16 * KC];    // 4 KB per wave (row-major)

  const int tid  = threadIdx.x;
  const int wave = tid >> 5;
  const int lane = tid & 31;

  const int nrb = (N + 15) >> 4;            // 16-row blocks
  int rb = blockIdx.x * WAVES + wave;
  const bool active = (rb < nrb);
  if (rb >= nrb) rb = nrb - 1;              // clamp; keep wave in barriers
  const int row0 = rb * 16;

  v8f cacc[8] = {};

  for (int kc = 0; kc < F_DIM; kc += KC) {
    __syncthreads();
    // Stage B chunk interleaved by K-pairs: 32 pair-rows x 32 float4-cols.
    {
      const float4* srcv = (const float4*)(kern + (size_t)kc * U_DIM);
      #pragma unroll
      for (int i = 0; i < 4; ++i) {
        const int idx = tid + 256 * i;      // 0..1023
        const int kp  = idx >> 5;           // K pair 0..31
        const int c4  = idx & 31;           // float4 column 0..31
        const float4 r0 = srcv[(2 * kp)     * 32 + c4];
        const float4 r1 = srcv[(2 * kp + 1) * 32 + c4];
        float2* dst = (float2*)&ldsB[(kp * U_DIM + c4 * 4) * 2];
        dst[0] = make_float2(r0.x, r1.x);
        dst[1] = make_float2(r0.y, r1.y);
        dst[2] = make_float2(r0.z, r1.z);
        dst[3] = make_float2(r0.w, r1.w);
      }
    }
    // Stage this wave's A chunk: 16 rows x 64 cols -> 256 float4, 8 per lane.
    {
      float4* dstv = (float4*)ldsA[wave];
      #pragma unroll
      for (int i = 0; i < 8; ++i) {
        const int idx = lane + 32 * i;      // 0..255
        const int r   = idx >> 4;           // row in block
        const int c4  = idx & 15;           // float4 col
        int grow = row0 + r; if (grow >= N) grow = N - 1;
        dstv[idx] = ((const float4*)(ns + (size_t)grow * F_DIM + kc))[c4];
      }
    }
    __syncthreads();

    const int m  = lane & 15;               // M for A / N-col for B
    const int kh = (lane >> 4) * 2;         // lane-half K offset (ISA layout)

    #pragma unroll
    for (int ks = 0; ks < KC; ks += 4) {
      // A 16x4 frag: v0={K+kh}, v1={K+kh+1}; contiguous -> one b64 load
      const v2f a = *(const v2f*)&ldsA[wave][m * KC + ks + kh];
      #pragma unroll
      for (int t = 0; t < 8; ++t) {
        const int col = t * 16 + m;
        // B 4x16 frag: K-pair interleaved -> one aligned b64 load
        const v2f b = *(const v2f*)&ldsB[(((ks + kh) >> 1) * U_DIM + col) * 2];
        cacc[t] = __builtin_amdgcn_wmma_f32_16x16x4_f32(
            false, a, false, b, (short)0, cacc[t], false, false);
      }
    }
  }

  if (active) {
    const int m    = lane & 15;
    const int half = lane >> 4;
    #pragma unroll
    for (int g = 0; g < 8; ++g) {           // C: VGPR g -> M = g + 8*half
      const int row = row0 + g + 8 * half;
      if (row < N) {
        float* orow = h + (size_t)row * U_DIM;
        #pragma unroll
        for (int t = 0; t < 8; ++t)
          orow[t * 16 + m] = cacc[t][g];
      }
    }
  }
}

// -------------------------------------------------------------------------
// Kernel 2: a_src[i] = h[i,:] . ka[0:128],  a_dst[i] = h[i,:] . ka[128:256]
// One wave32 per node; float4 loads + shfl_xor reduction.
// -------------------------------------------------------------------------
__global__ __launch_bounds__(256) void gat_attn_vec(
    const float* __restrict__ h, const float* __restrict__ ka,
    float* __restrict__ a_src, float* __restrict__ a_dst, int N)
{
  const int node = (int)((blockIdx.x * blockDim.x + threadIdx.x) >> 5);
  const int lane = threadIdx.x & 31;
  if (node >= N) return;

  const float4 hv = ((const float4*)(h + (size_t)node * U_DIM))[lane];
  const float4 kl = ((const float4*)ka)[lane];
  const float4 kh = ((const float4*)(ka + U_DIM))[lane];

  float s = hv.x * kl.x + hv.y * kl.y + hv.z * kl.z + hv.w * kl.w;
  float d = hv.x * kh.x + hv.y * kh.y + hv.z * kh.z + hv.w * kh.w;
  #pragma unroll
  for (int off = 16; off > 0; off >>= 1) {
    s += __shfl_xor(s, off, 32);
    d += __shfl_xor(d, off, 32);
  }
  if (lane == 0) { a_src[node] = s; a_dst[node] = d; }
}

// -------------------------------------------------------------------------
// Kernel 3: per-node softmax over its 16 contiguous edges + weighted
// aggregation of gathered h[dst] rows. One wave32 per node.
// The 819 MB random row gather is the roofline term (~35us @ 23.3 TB/s):
// prefetch rows early, then keep all 16 coalesced 512B row loads in flight.
// -------------------------------------------------------------------------
__global__ __launch_bounds__(256) void gat_aggregate(
    const float* __restrict__ h, const int* __restrict__ edges,
    const float* __restrict__ ew, const float* __restrict__ a_src,
    const float* __restrict__ a_dst, float* __restrict__ out, int N)
{
  const int node = (int)((blockIdx.x * blockDim.x + threadIdx.x) >> 5);
  const int lane = threadIdx.x & 31;
  if (node >= N) return;

  int   dd = 0;
  float sc = 0.0f;
  float w  = 0.0f;
  if (lane < DEG) {
    const int j = node * DEG + lane;
    dd = edges[2 * j + 1];                          // dst node
    w  = ew[j];
  }
  // Early prefetch of gather rows: lanes 0-15 cover row starts, lanes 16-31
  // cover the second 256B half of the same rows (global_prefetch_b8).
  {
    const int ddp = __shfl(dd, lane & 15, 32);
    const float* p = h + (size_t)ddp * U_DIM + ((lane >> 4) ? 64 : 0);
    __builtin_prefetch((const void*)p, 0, 3);
  }
  if (lane < DEG) {
    float s = w * (a_src[node] + a_dst[dd]);
    s = (s > 0.0f) ? s : 0.2f * s;                  // leaky_relu, alpha=0.2
    s = fminf(2.0f, fmaxf(-2.0f, s));               // clip
    sc = __expf(s);
  }
  float ssum = sc;                                  // lanes >=16 contribute 0
  #pragma unroll
  for (int off = 16; off > 0; off >>= 1) ssum += __shfl_xor(ssum, off, 32);
  const float norm = sc / ssum;

  // Broadcast all 16 (dst, weight) pairs, then issue all 16 row loads
  // before accumulating -> 16 independent 512B coalesced loads in flight.
  int   dks[DEG];
  float nks[DEG];
  #pragma unroll
  for (int k = 0; k < DEG; ++k) {
    dks[k] = __shfl(dd,   k, 32);
    nks[k] = __shfl(norm, k, 32);
  }
  float4 hv[DEG];
  #pragma unroll
  for (int k = 0; k < DEG; ++k)
    hv[k] = ((const float4*)(h + (size_t)dks[k] * U_DIM))[lane];

  float4 acc = {0.0f, 0.0f, 0.0f, 0.0f};
  #pragma unroll
  for (int k = 0; k < DEG; ++k) {
    acc.x += nks[k] * hv[k].x; acc.y += nks[k] * hv[k].y;
    acc.z += nks[k] * hv[k].z; acc.w += nks[k] * hv[k].w;
  }
  ((float4*)(out + (size_t)node * U_DIM))[lane] = acc;
}

// -------------------------------------------------------------------------
extern "C" void kernel_launch(void* const* d_in, const int* in_sizes, int n_in,
                              void* d_out, int out_size, void* d_ws, size_t ws_size,
                              hipStream_t stream) {
  const float* ns    = (const float*)d_in[0];   // [1,N,256] f32
  const int*   edges = (const int*)  d_in[1];   // [1,E,2]  i32
  const float* ew    = (const float*)d_in[2];   // [1,E]    f32
  const float* kern  = (const float*)d_in[3];   // [256,128] f32
  const float* ka    = (const float*)d_in[4];   // [256,1]  f32
  float*       out   = (float*)d_out;           // [N,128]  f32

  const int N = in_sizes[0] / F_DIM;

  // Workspace carve-out: h[N*128] | a_src[N] | a_dst[N]
  float* h     = (float*)d_ws;
  float* a_src = h + (size_t)N * U_DIM;
  float* a_dst = a_src + N;

  const int nrb         = (N + 15) / 16;
  const int gemm_blocks = (nrb + WAVES - 1) / WAVES;
  gat_gemm_h<<<gemm_blocks, 256, 0, stream>>>(ns, kern, h, N);

  const int wave_blocks = (N + 7) / 8;          // 8 wave32 per 256-thread block
  gat_attn_vec<<<wave_blocks, 256, 0, stream>>>(h, ka, a_src, a_dst, N);
  gat_aggregate<<<wave_blocks, 256, 0, stream>>>(h, edges, ew, a_src, a_dst,
                                                 out, N);
}